// DiscriminatorModel_76965813945077
// MI455X (gfx1250) — compile-verified
//
#include <hip/hip_runtime.h>
#include <hip/hip_bf16.h>

typedef float v2f __attribute__((ext_vector_type(2)));
typedef float v8f __attribute__((ext_vector_type(8)));

#define LN_EPS 1e-5f
#define SLOPE 0.2f
#define WAVES 4
#define ROWS_PER_BLOCK 64   // 4 waves x 16 rows
#define HS1 34              // padded LDS row stride for 32-wide activations
#define HS2 66              // padded LDS row stride for 64-wide activations

__device__ __forceinline__ v8f wmma4(v2f a, v2f b, v8f c) {
  // D = A(16x4 f32) * B(4x16 f32) + C(16x16 f32)
  return __builtin_amdgcn_wmma_f32_16x16x4_f32(false, a, false, b, (short)0, c,
                                               false, false);
}

// butterfly sum across each 16-lane half of the wave32
__device__ __forceinline__ float hsum16(float v) {
  v += __shfl_xor(v, 1, 32);
  v += __shfl_xor(v, 2, 32);
  v += __shfl_xor(v, 4, 32);
  v += __shfl_xor(v, 8, 32);
  return v;
}

template <int N>
__device__ __forceinline__ void ln_lrelu(float* h, const float* __restrict__ g,
                                         const float* __restrict__ b) {
  float s = 0.f;
#pragma unroll
  for (int i = 0; i < N; ++i) s += h[i];
  float m = s * (1.f / N);
  float s2 = 0.f;
#pragma unroll
  for (int i = 0; i < N; ++i) { float d = h[i] - m; s2 += d * d; }
  float r = rsqrtf(s2 * (1.f / N) + LN_EPS);
#pragma unroll
  for (int i = 0; i < N; ++i) {
    float y = (h[i] - m) * r * g[i] + b[i];
    h[i] = y > 0.f ? y : SLOPE * y;
  }
}

__global__ __launch_bounds__(128) void disc_mlp_kernel(
    const float* __restrict__ x,
    const float* __restrict__ W1, const float* __restrict__ g1, const float* __restrict__ bt1,
    const float* __restrict__ W2, const float* __restrict__ g2, const float* __restrict__ bt2,
    const float* __restrict__ W3, const float* __restrict__ g3, const float* __restrict__ bt3,
    const float* __restrict__ W4, const float* __restrict__ g4, const float* __restrict__ bt4,
    const float* __restrict__ W5, const float* __restrict__ g5, const float* __restrict__ bt5,
    const float* __restrict__ W6, const float* __restrict__ g6, const float* __restrict__ bt6,
    const float* __restrict__ W7, const float* __restrict__ g7, const float* __restrict__ bt7,
    const float* __restrict__ W8, const float* __restrict__ b8,
    float* __restrict__ out, int nRows) {
  // Weights packed as consecutive-k float2 pairs: sWp[(k>>1)*N*2 + n*2 + (k&1)]
  __shared__ __align__(16) float sW1p[128 * 32];
  __shared__ __align__(16) float sW2p[32 * 64];
  __shared__ __align__(16) float sW3p[64 * 32];
  __shared__ __align__(16) float sH[WAVES][16 * HS2];  // per-wave activation tile

  const int tid = threadIdx.x;
  for (int i = tid; i < 128 * 32; i += 128) {
    int k = i >> 5, n = i & 31;
    sW1p[((k >> 1) * 32 + n) * 2 + (k & 1)] = W1[i];
  }
  for (int i = tid; i < 32 * 64; i += 128) {
    int k = i >> 6, n = i & 63;
    sW2p[((k >> 1) * 64 + n) * 2 + (k & 1)] = W2[i];
  }
  for (int i = tid; i < 64 * 32; i += 128) {
    int k = i >> 5, n = i & 31;
    sW3p[((k >> 1) * 32 + n) * 2 + (k & 1)] = W3[i];
  }
  __syncthreads();

  const int lane = tid & 31;
  const int half = lane >> 4;  // which 16-lane half
  const int n16 = lane & 15;   // column / row-in-tile index
  const int wave = tid >> 5;
  float* hb = &sH[wave][0];
  const long rowBase = (long)blockIdx.x * ROWS_PER_BLOCK + wave * 16;

  // clamp row for loads; WMMA requires full EXEC so no divergence here
  long myRow = rowBase + n16;
  long ldRow = myRow < nRows ? myRow : (long)nRows - 1;

  // ---------------- Layer 1: x(16x128) @ W1(128x32) ----------------
  const float* xrow = x + ldRow * 128;
  v8f c0 = {}, c1 = {};
#pragma unroll 8
  for (int k0 = 0; k0 < 128; k0 += 4) {
    v2f a = *(const v2f*)(xrow + k0 + 2 * half);          // A layout
    const float* bp = sW1p + ((k0 >> 1) + half) * 64;     // pair-row of W1
    v2f b0 = *(const v2f*)(bp + n16 * 2);                 // N = 0..15
    v2f b1 = *(const v2f*)(bp + 32 + n16 * 2);            // N = 16..31
    c0 = wmma4(a, b0, c0);
    c1 = wmma4(a, b1, c1);
  }

  // LN + LeakyReLU over 32 features (row = v + 8*half, col = n16 [+16])
  {
    float ga = g1[n16], gb = g1[n16 + 16];
    float ba = bt1[n16], bb = bt1[n16 + 16];
#pragma unroll
    for (int v = 0; v < 8; ++v) {
      float p = c0[v], q = c1[v];
      float s = hsum16(p + q);
      float s2 = hsum16(p * p + q * q);
      float m = s * (1.f / 32.f);
      float r = rsqrtf(s2 * (1.f / 32.f) - m * m + LN_EPS);
      float h0 = (p - m) * r * ga + ba; h0 = h0 > 0.f ? h0 : SLOPE * h0;
      float h1 = (q - m) * r * gb + bb; h1 = h1 > 0.f ? h1 : SLOPE * h1;
      int mr = v + 8 * half;
      hb[mr * HS1 + n16] = h0;
      hb[mr * HS1 + n16 + 16] = h1;
    }
  }

  // ---------------- Layer 2: h1(16x32) @ W2(32x64) ----------------
  v8f d0 = {}, d1 = {}, d2 = {}, d3 = {};
  {
    const float* ar = hb + n16 * HS1;
#pragma unroll
    for (int k0 = 0; k0 < 32; k0 += 4) {
      v2f a = *(const v2f*)(ar + k0 + 2 * half);
      const float* bp = sW2p + ((k0 >> 1) + half) * 128;
      d0 = wmma4(a, *(const v2f*)(bp + n16 * 2), d0);
      d1 = wmma4(a, *(const v2f*)(bp + 32 + n16 * 2), d1);
      d2 = wmma4(a, *(const v2f*)(bp + 64 + n16 * 2), d2);
      d3 = wmma4(a, *(const v2f*)(bp + 96 + n16 * 2), d3);
    }
  }
  {
    float ga0 = g2[n16], ga1 = g2[n16 + 16], ga2 = g2[n16 + 32], ga3 = g2[n16 + 48];
    float ba0 = bt2[n16], ba1 = bt2[n16 + 16], ba2 = bt2[n16 + 32], ba3 = bt2[n16 + 48];
#pragma unroll
    for (int v = 0; v < 8; ++v) {
      float p0 = d0[v], p1 = d1[v], p2 = d2[v], p3 = d3[v];
      float s = hsum16(p0 + p1 + p2 + p3);
      float s2 = hsum16(p0 * p0 + p1 * p1 + p2 * p2 + p3 * p3);
      float m = s * (1.f / 64.f);
      float r = rsqrtf(s2 * (1.f / 64.f) - m * m + LN_EPS);
      int mr = v + 8 * half;
      float t;
      t = (p0 - m) * r * ga0 + ba0; hb[mr * HS2 + n16]      = t > 0.f ? t : SLOPE * t;
      t = (p1 - m) * r * ga1 + ba1; hb[mr * HS2 + n16 + 16] = t > 0.f ? t : SLOPE * t;
      t = (p2 - m) * r * ga2 + ba2; hb[mr * HS2 + n16 + 32] = t > 0.f ? t : SLOPE * t;
      t = (p3 - m) * r * ga3 + ba3; hb[mr * HS2 + n16 + 48] = t > 0.f ? t : SLOPE * t;
    }
  }

  // ---------------- Layer 3: h2(16x64) @ W3(64x32) ----------------
  v8f e0 = {}, e1 = {};
  {
    const float* ar = hb + n16 * HS2;
#pragma unroll
    for (int k0 = 0; k0 < 64; k0 += 4) {
      v2f a = *(const v2f*)(ar + k0 + 2 * half);
      const float* bp = sW3p + ((k0 >> 1) + half) * 64;
      e0 = wmma4(a, *(const v2f*)(bp + n16 * 2), e0);
      e1 = wmma4(a, *(const v2f*)(bp + 32 + n16 * 2), e1);
    }
  }
  {
    float ga = g3[n16], gb = g3[n16 + 16];
    float ba = bt3[n16], bb = bt3[n16 + 16];
#pragma unroll
    for (int v = 0; v < 8; ++v) {
      float p = e0[v], q = e1[v];
      float s = hsum16(p + q);
      float s2 = hsum16(p * p + q * q);
      float m = s * (1.f / 32.f);
      float r = rsqrtf(s2 * (1.f / 32.f) - m * m + LN_EPS);
      float h0 = (p - m) * r * ga + ba; h0 = h0 > 0.f ? h0 : SLOPE * h0;
      float h1 = (q - m) * r * gb + bb; h1 = h1 > 0.f ? h1 : SLOPE * h1;
      int mr = v + 8 * half;
      hb[mr * HS1 + n16] = h0;
      hb[mr * HS1 + n16 + 16] = h1;
    }
  }

  // ---------------- Tail layers 4..8: one row per lane (lanes 0-15) ----------------
  if (half == 0) {
    float h[32];
#pragma unroll
    for (int j = 0; j < 32; ++j) h[j] = hb[n16 * HS1 + j];

    float t4[16];
#pragma unroll
    for (int j = 0; j < 16; ++j) {
      float s = 0.f;
#pragma unroll
      for (int k = 0; k < 32; ++k) s += h[k] * W4[k * 16 + j];
      t4[j] = s;
    }
    ln_lrelu<16>(t4, g4, bt4);

    float t5[8];
#pragma unroll
    for (int j = 0; j < 8; ++j) {
      float s = 0.f;
#pragma unroll
      for (int k = 0; k < 16; ++k) s += t4[k] * W5[k * 8 + j];
      t5[j] = s;
    }
    ln_lrelu<8>(t5, g5, bt5);

    float t6[4];
#pragma unroll
    for (int j = 0; j < 4; ++j) {
      float s = 0.f;
#pragma unroll
      for (int k = 0; k < 8; ++k) s += t5[k] * W6[k * 4 + j];
      t6[j] = s;
    }
    ln_lrelu<4>(t6, g6, bt6);

    float t7[2];
#pragma unroll
    for (int j = 0; j < 2; ++j) {
      float s = 0.f;
#pragma unroll
      for (int k = 0; k < 4; ++k) s += t6[k] * W7[k * 2 + j];
      t7[j] = s;
    }
    ln_lrelu<2>(t7, g7, bt7);

    if (myRow < nRows)
      out[myRow] = t7[0] * W8[0] + t7[1] * W8[1] + b8[0];
  }
}

extern "C" void kernel_launch(void* const* d_in, const int* in_sizes, int n_in,
                              void* d_out, int out_size, void* d_ws, size_t ws_size,
                              hipStream_t stream) {
  (void)n_in; (void)out_size; (void)d_ws; (void)ws_size;
  const float* x   = (const float*)d_in[0];
  const float* W1  = (const float*)d_in[1];
  const float* g1  = (const float*)d_in[2];
  const float* bt1 = (const float*)d_in[3];
  const float* W2  = (const float*)d_in[4];
  const float* g2  = (const float*)d_in[5];
  const float* bt2 = (const float*)d_in[6];
  const float* W3  = (const float*)d_in[7];
  const float* g3  = (const float*)d_in[8];
  const float* bt3 = (const float*)d_in[9];
  const float* W4  = (const float*)d_in[10];
  const float* g4  = (const float*)d_in[11];
  const float* bt4 = (const float*)d_in[12];
  const float* W5  = (const float*)d_in[13];
  const float* g5  = (const float*)d_in[14];
  const float* bt5 = (const float*)d_in[15];
  const float* W6  = (const float*)d_in[16];
  const float* g6  = (const float*)d_in[17];
  const float* bt6 = (const float*)d_in[18];
  const float* W7  = (const float*)d_in[19];
  const float* g7  = (const float*)d_in[20];
  const float* bt7 = (const float*)d_in[21];
  const float* W8  = (const float*)d_in[22];
  const float* b8  = (const float*)d_in[23];

  int nRows = in_sizes[0] / 128;
  int grid = (nRows + ROWS_PER_BLOCK - 1) / ROWS_PER_BLOCK;
  if (grid <= 0) return;
  disc_mlp_kernel<<<grid, 128, 0, stream>>>(
      x, W1, g1, bt1, W2, g2, bt2, W3, g3, bt3, W4, g4, bt4,
      W5, g5, bt5, W6, g6, bt6, W7, g7, bt7, W8, b8,
      (float*)d_out, nRows);
}